// GNN_43379169689765
// MI455X (gfx1250) — compile-verified
//
#include <hip/hip_runtime.h>
#include <hip/hip_bf16.h>

typedef __attribute__((ext_vector_type(2))) float v2f;
typedef __attribute__((ext_vector_type(8))) float v8f;

#define DIM 64
#define LEAKY 0.01f

// ---------------------------------------------------------------------------
// z = (relu?)(hin) @ W^T + b   via V_WMMA_F32_16X16X4_F32
// Block = 128 threads (4 waves). Each wave computes a 16x64 output tile.
// W (64x64 f32, row-major, 16KB) is staged in LDS once per block.
// A layout (16x4 f32): lane<16 -> M=lane,   K = 4t+0 (v0), 4t+1 (v1)
//                      lane>=16 -> M=lane-16, K = 4t+2 (v0), 4t+3 (v1)
// B layout (4x16 f32): lane<16 -> N=lane,   K = 4t+0 (v0), 4t+1 (v1)  [B[k][n]=W[n][k]]
//                      lane>=16 -> N=lane-16, K = 4t+2, 4t+3
// C/D layout (16x16 f32): vgpr r, lane<16 -> (M=r,   N=lane)
//                                  lane>=16 -> (M=r+8, N=lane-16)
// ---------------------------------------------------------------------------
__global__ __launch_bounds__(128)
void gat_gemm_wmma(const float* __restrict__ hin, const float* __restrict__ W,
                   const float* __restrict__ bias, float* __restrict__ z,
                   int nTiles, int reluIn) {
  __shared__ float sW[DIM * DIM];
  __shared__ float sB[DIM];

  const int tid = threadIdx.x;
  for (int i = tid; i < DIM * DIM; i += 128) sW[i] = W[i];
  if (tid < DIM) sB[tid] = bias[tid];
  __syncthreads();

  const int wave = tid >> 5;
  const int lane = tid & 31;
  const int rowTile = blockIdx.x * 4 + wave;
  if (rowTile >= nTiles) return;

  const int row0 = rowTile * 16;
  const int laneRow = lane & 15;           // M (or N) within tile
  const int kHalf = (lane >> 4) * 2;       // 0 or 2

  // Load this wave's A block: row (row0+laneRow), 16 k-tiles of 2 floats each.
  const float* arow = hin + (size_t)(row0 + laneRow) * DIM;
  v2f aR[16];
#pragma unroll
  for (int t = 0; t < 16; ++t) {
    v2f a = *(const v2f*)(arow + 4 * t + kHalf);
    if (reluIn) { a.x = fmaxf(a.x, 0.0f); a.y = fmaxf(a.y, 0.0f); }
    aR[t] = a;
  }

  const int rOff = (lane < 16) ? 0 : 8;
#pragma unroll
  for (int nt = 0; nt < 4; ++nt) {
    const int n0 = nt * 16;
    const float bcol = sB[n0 + laneRow];
    v8f acc;
#pragma unroll
    for (int r = 0; r < 8; ++r) acc[r] = bcol;   // C init = bias (per-column)

#pragma unroll
    for (int t = 0; t < 16; ++t) {
      // B[k][n] = W[n][k]; this lane: n = n0+laneRow, k = 4t+kHalf (+1)
      v2f b = *(const v2f*)(&sW[(n0 + laneRow) * DIM + 4 * t + kHalf]);
      acc = __builtin_amdgcn_wmma_f32_16x16x4_f32(
          /*neg_a=*/false, aR[t], /*neg_b=*/false, b,
          /*c_mod=*/(short)0, acc, /*reuse_a=*/false, /*reuse_b=*/false);
    }

    float* zr = z + (size_t)(row0 + rOff) * DIM + n0 + laneRow;
#pragma unroll
    for (int r = 0; r < 8; ++r) zr[(size_t)r * DIM] = acc[r];
  }
}

// ---------------------------------------------------------------------------
// Per-node attention scalars: s_src[n] = z[n]·a[0:64], s_dst[n] = z[n]·a[64:128]
// One wave per node; lane k handles channels k and k+32; wave32 xor-reduce.
// ---------------------------------------------------------------------------
__global__ __launch_bounds__(256)
void node_scores(const float* __restrict__ z, const float* __restrict__ a,
                 float* __restrict__ ssrc, float* __restrict__ sdst, int nN) {
  const int lane = threadIdx.x & 31;
  const int warp = (blockIdx.x * blockDim.x + threadIdx.x) >> 5;
  const int nWarps = (gridDim.x * blockDim.x) >> 5;

  const float a0 = a[lane], a1 = a[lane + 32];
  const float a2 = a[lane + 64], a3 = a[lane + 96];

  for (int n = warp; n < nN; n += nWarps) {
    const float* zr = z + (size_t)n * DIM;
    const float z0 = zr[lane], z1 = zr[lane + 32];
    float ps = z0 * a0 + z1 * a1;
    float pd = z0 * a2 + z1 * a3;
#pragma unroll
    for (int off = 16; off > 0; off >>= 1) {
      ps += __shfl_xor(ps, off, 32);
      pd += __shfl_xor(pd, off, 32);
    }
    if (lane == 0) { ssrc[n] = ps; sdst[n] = pd; }
  }
}

__device__ __forceinline__ unsigned fmap(float f) {
  unsigned u = __float_as_uint(f);
  return (u & 0x80000000u) ? ~u : (u | 0x80000000u);
}
__device__ __forceinline__ float funmap(unsigned u) {
  return (u & 0x80000000u) ? __uint_as_float(u ^ 0x80000000u) : __uint_as_float(~u);
}

// Pass 1: e = leaky_relu(s_src[src]+s_dst[dst]+ab); store e; segment-max into mbits[dst]
__global__ __launch_bounds__(256)
void edge_scores_max(const int* __restrict__ src, const int* __restrict__ dst,
                     const float* __restrict__ ssrc, const float* __restrict__ sdst,
                     const float* __restrict__ abp, float* __restrict__ ebuf,
                     unsigned* __restrict__ mbits, int nE) {
  const float ab = abp[0];
  for (int e = blockIdx.x * blockDim.x + threadIdx.x; e < nE;
       e += gridDim.x * blockDim.x) {
    const int d = dst[e];
    float v = ssrc[src[e]] + sdst[d] + ab;
    v = (v > 0.0f) ? v : LEAKY * v;
    ebuf[e] = v;
    atomicMax(&mbits[d], fmap(v));
  }
}

// Pass 2: ex = exp(e - m[dst]); store ex; segment-sum into den[dst]
__global__ __launch_bounds__(256)
void edge_exp_sum(const int* __restrict__ dst, const unsigned* __restrict__ mbits,
                  float* __restrict__ ebuf, float* __restrict__ den, int nE) {
  for (int e = blockIdx.x * blockDim.x + threadIdx.x; e < nE;
       e += gridDim.x * blockDim.x) {
    const int d = dst[e];
    const float ex = __expf(ebuf[e] - funmap(mbits[d]));
    ebuf[e] = ex;
    atomicAdd(&den[d], ex);
  }
}

// Pass 3: out[dst] += (ex/den[dst]) * z[src]   (warp per edge; lane = channel)
__global__ __launch_bounds__(256)
void edge_aggregate(const int* __restrict__ src, const int* __restrict__ dst,
                    const float* __restrict__ ebuf, const float* __restrict__ den,
                    const float* __restrict__ z, float* __restrict__ out, int nE) {
  const int lane = threadIdx.x & 31;
  const int warp = (blockIdx.x * blockDim.x + threadIdx.x) >> 5;
  const int nWarps = (gridDim.x * blockDim.x) >> 5;

  for (int e = warp; e < nE; e += nWarps) {
    const int s = src[e], d = dst[e];
    const float alpha = ebuf[e] / den[d];
    const float* zr = z + (size_t)s * DIM;
    float* orow = out + (size_t)d * DIM;
    atomicAdd(orow + lane,      alpha * zr[lane]);
    atomicAdd(orow + lane + 32, alpha * zr[lane + 32]);
  }
}

extern "C" void kernel_launch(void* const* d_in, const int* in_sizes, int n_in,
                              void* d_out, int out_size, void* d_ws, size_t ws_size,
                              hipStream_t stream) {
  const float* x  = (const float*)d_in[0];
  const int* src  = (const int*)d_in[1];
  const int* dst  = (const int*)d_in[2];
  const int nN = in_sizes[0] / DIM;
  const int nE = in_sizes[1];

  char* ws = (char*)d_ws;
  const size_t featB = (size_t)nN * DIM * sizeof(float);
  float*    z     = (float*)ws;
  float*    h     = (float*)(ws + featB);
  float*    ebuf  = (float*)(ws + 2 * featB);
  unsigned* mbits = (unsigned*)(ws + 2 * featB + (size_t)nE * 4);
  float*    den   = (float*)(ws + 2 * featB + (size_t)nE * 4 + (size_t)nN * 4);
  float*    ssrc  = (float*)(ws + 2 * featB + (size_t)nE * 4 + (size_t)nN * 8);
  float*    sdst  = (float*)(ws + 2 * featB + (size_t)nE * 4 + (size_t)nN * 12);

  const int nTiles = (nN + 15) / 16;
  const dim3 gemmGrid((nTiles + 3) / 4), gemmBlk(128);
  const int edgeBlocks = (nE + 255) / 256;
  const int nodeBlocks = ((nN + 7) / 8);   // 8 warps/block, warp per node
  const int aggBlocks = 8192;              // grid-stride, warp per edge

  for (int L = 0; L < 3; ++L) {
    const float* W  = (const float*)d_in[3 + 4 * L];
    const float* b  = (const float*)d_in[4 + 4 * L];
    const float* a  = (const float*)d_in[5 + 4 * L];
    const float* ab = (const float*)d_in[6 + 4 * L];

    const float* hin = (L == 0) ? x : h;
    float* aggout = (L == 2) ? (float*)d_out : h;
    const int reluIn = (L > 0) ? 1 : 0;

    // z = (relu?)(hin) @ W^T + b   (WMMA)
    gat_gemm_wmma<<<gemmGrid, gemmBlk, 0, stream>>>(hin, W, b, z, nTiles, reluIn);

    // per-node attention scalars
    node_scores<<<nodeBlocks, 256, 0, stream>>>(z, a, ssrc, sdst, nN);

    // zero segment buffers + aggregation target (after gemm consumed h)
    hipMemsetAsync(mbits, 0, (size_t)nN * 4, stream);  // 0 == fmap(-NaN), identity for max
    hipMemsetAsync(den,   0, (size_t)nN * 4, stream);
    hipMemsetAsync(aggout, 0, featB, stream);

    edge_scores_max<<<edgeBlocks, 256, 0, stream>>>(src, dst, ssrc, sdst, ab,
                                                    ebuf, mbits, nE);
    edge_exp_sum<<<edgeBlocks, 256, 0, stream>>>(dst, mbits, ebuf, den, nE);
    edge_aggregate<<<aggBlocks, 256, 0, stream>>>(src, dst, ebuf, den, z, aggout, nE);
  }
}